// SoftMarginRankLoss_30940944401148
// MI455X (gfx1250) — compile-verified
//
#include <hip/hip_runtime.h>

#define NT   256        // threads per block (8 wave32)
#define TOPK 30
#define CC   32000      // columns
#define NF4  (CC / 4)   // 8000 float4 per row
#define NB   4096       // rows
#define CAND 512        // candidate buffer cap

// Map float -> unsigned so that unsigned ascending == float ascending.
__device__ __forceinline__ unsigned int mapf(float f) {
    unsigned int u = __float_as_uint(f);
    return (u & 0x80000000u) ? ~u : (u | 0x80000000u);
}

struct SharedBlk {
    float row[CC] __attribute__((aligned(16)));   // 128000 B staged row (async DMA target)
    unsigned int hist[4096];                      // 12-bit radix histogram (reused as 256-bin)
    unsigned int coarse[NT];                      // coarse sums for fast descending scan
    unsigned int candU[CAND];
    int          candI[CAND];
    unsigned long long red[8];                    // per-wave partial argmax
    unsigned long long win;                       // broadcast winner key
    int   selT[TOPK];
    int   selP[TOPK];
    float loss[TOPK];
    unsigned int kk;                              // remaining rank within prefix bucket
    unsigned int pfx;                             // radix prefix
    unsigned int candCnt;
    unsigned int ovl;
};

// Select top-TOPK indices (jax.lax.top_k semantics: descending value, ties -> lower
// index) of one row into selOut. Leaves the row resident in sh->row.
__device__ void topk_idx(const float* __restrict__ rowPtr, SharedBlk* sh,
                         int* selOut, int tid) {
    // ---- Issue async DMA of the whole row into LDS (overlaps with pass 1) ----
    unsigned int ldsBase = (unsigned int)(uintptr_t)(&sh->row[0]);
    for (int i = tid; i < NF4; i += NT) {
        unsigned int lo = ldsBase + (unsigned int)(i * 16); // LDS byte address
        unsigned int go = (unsigned int)(i * 16);           // global byte offset (GVS)
        asm volatile("global_load_async_to_lds_b128 %0, %1, %2"
                     :: "v"(lo), "v"(go), "s"(rowPtr)
                     : "memory");
    }

    // ---- Init ----
    for (int i = tid; i < 4096; i += NT) sh->hist[i] = 0u;
    if (tid == 0) { sh->kk = TOPK; sh->candCnt = 0u; }
    __syncthreads();

    // ---- Pass 1: 12-bit histogram, streamed from GLOBAL (while DMA fills LDS) ----
    const float4* g4 = (const float4*)rowPtr;
    for (int i = tid; i < NF4; i += NT) {
        float4 v = g4[i];
        atomicAdd(&sh->hist[mapf(v.x) >> 20], 1u);
        atomicAdd(&sh->hist[mapf(v.y) >> 20], 1u);
        atomicAdd(&sh->hist[mapf(v.z) >> 20], 1u);
        atomicAdd(&sh->hist[mapf(v.w) >> 20], 1u);
    }
    __syncthreads();

    // coarse sums (16 bins each) so thread 0 scans 256+16 instead of 4096
    unsigned int cs = 0u;
    for (int b = 0; b < 16; ++b) cs += sh->hist[tid * 16 + b];
    sh->coarse[tid] = cs;
    __syncthreads();

    if (tid == 0) {
        unsigned int kk = sh->kk, cum = 0u; int bin = 4095;
        for (int c = 255; c >= 0; --c) {
            if (cum + sh->coarse[c] >= kk) {
                for (int b = c * 16 + 15; b >= c * 16; --b) {
                    unsigned int h = sh->hist[b];
                    if (cum + h >= kk) { bin = b; break; }
                    cum += h;
                }
                break;
            }
            cum += sh->coarse[c];
        }
        sh->kk  = kk - cum;            // rank within chosen 12-bit bucket
        sh->pfx = (unsigned int)bin;   // 12-bit prefix
    }
    __syncthreads();

    // zero 256-bin histogram for pass 2; wait for the async DMA to land
    sh->hist[tid] = 0u;
    asm volatile("s_wait_asynccnt 0" ::: "memory");
    __syncthreads();

    unsigned int p12 = sh->pfx;

    // ---- Pass 2: refine next 8 bits, reading the row from LDS ----
    const float4* s4 = (const float4*)sh->row;
    for (int i = tid; i < NF4; i += NT) {
        float4 v = s4[i];
        unsigned int u;
        u = mapf(v.x); if ((u >> 20) == p12) atomicAdd(&sh->hist[(u >> 12) & 0xFFu], 1u);
        u = mapf(v.y); if ((u >> 20) == p12) atomicAdd(&sh->hist[(u >> 12) & 0xFFu], 1u);
        u = mapf(v.z); if ((u >> 20) == p12) atomicAdd(&sh->hist[(u >> 12) & 0xFFu], 1u);
        u = mapf(v.w); if ((u >> 20) == p12) atomicAdd(&sh->hist[(u >> 12) & 0xFFu], 1u);
    }
    __syncthreads();

    if (tid == 0) {
        unsigned int kk = sh->kk, cum = 0u; int bin = 0;
        for (int b = 255; b >= 0; --b) {
            unsigned int h = sh->hist[b];
            if (cum + h >= kk) { bin = b; break; }
            cum += h;
        }
        sh->kk  = kk - cum;
        sh->pfx = (p12 << 8) | (unsigned int)bin;  // 20-bit prefix
        sh->win = 0ull;
    }
    __syncthreads();

    unsigned int p20 = sh->pfx;

    // ---- Collect all elements with (u>>12) >= 20-bit prefix (superset of top-K) ----
    for (int i = tid; i < NF4; i += NT) {
        float4 v = s4[i];
        float e[4] = {v.x, v.y, v.z, v.w};
        #pragma unroll
        for (int c = 0; c < 4; ++c) {
            unsigned int u = mapf(e[c]);
            if ((u >> 12) >= p20) {
                unsigned int p = atomicAdd(&sh->candCnt, 1u);
                if (p < CAND) { sh->candU[p] = u; sh->candI[p] = i * 4 + c; }
            }
        }
    }
    __syncthreads();

    unsigned int cc = sh->candCnt;
    int nC = cc < (unsigned int)CAND ? (int)cc : CAND;

    // ---- Exact top-K: 30 block-wide argmax extractions over ~40-60 candidates ----
    // key = (u << 32) | ~idx  : larger value wins, ties -> smaller index wins.
    int lane = tid & 31;
    int wid  = tid >> 5;
    for (int s = 0; s < TOPK; ++s) {
        unsigned long long prev = sh->win;
        unsigned long long best = 0ull;
        for (int j = tid; j < nC; j += NT) {     // strided: each slot owned by one thread
            unsigned int u = sh->candU[j];
            unsigned long long key =
                u ? (((unsigned long long)u << 32) | (unsigned int)(~sh->candI[j])) : 0ull;
            if (key != 0ull && key == prev) { sh->candU[j] = 0u; key = 0ull; }  // remove prev winner
            best = key > best ? key : best;
        }
        // wave32 shuffle reduction
        for (int off = 16; off > 0; off >>= 1) {
            unsigned long long o = __shfl_xor(best, off, 32);
            best = o > best ? o : best;
        }
        if (lane == 0) sh->red[wid] = best;
        __syncthreads();
        if (tid == 0) {
            unsigned long long b = sh->red[0];
            for (int w = 1; w < 8; ++w) b = sh->red[w] > b ? sh->red[w] : b;
            sh->win = b;
            selOut[s] = (int)(~(unsigned int)b);
        }
        __syncthreads();
    }
}

__global__ __launch_bounds__(NT)
void SoftMarginRankLoss_row_kernel(const float* __restrict__ logits,
                                   const float* __restrict__ targets,
                                   float* __restrict__ ws) {
    __shared__ SharedBlk sh;
    const int row = blockIdx.x;
    const int tid = threadIdx.x;

    const float* tRow = targets + (size_t)row * CC;
    const float* lRow = logits  + (size_t)row * CC;

    topk_idx(tRow, &sh, sh.selT, tid);   // true_idx (targets)
    __syncthreads();
    topk_idx(lRow, &sh, sh.selP, tid);   // pred_idx (logits); logits row stays in LDS
    __syncthreads();

    if (tid == 0) sh.ovl = 0u;
    __syncthreads();

    if (tid < TOPK) {
        int ti = sh.selT[tid];
        ti = ti < 0 ? 0 : (ti >= CC ? CC - 1 : ti);
        float x = sh.row[ti];            // logits[true_idx] gathered from LDS
        int hit = 0;
        for (int m = 0; m < TOPK; ++m) hit |= (sh.selP[m] == sh.selT[tid]);
        if (hit) atomicAdd(&sh.ovl, 1u); // integer: deterministic
        float p = 1.0f / (1.0f + __expf(-x));
        sh.loss[tid] = -__logf(p + 1e-7f);
    }
    __syncthreads();

    if (tid == 0) {
        float acc = 0.0f;
        for (int j = 0; j < TOPK; ++j) acc += sh.loss[j];   // fixed order: deterministic
        float w = 1.0f - (float)sh.ovl / (float)TOPK;
        ws[row] = (acc / (float)TOPK) * w;
    }
}

__global__ __launch_bounds__(NT)
void SoftMarginRankLoss_reduce_kernel(const float* __restrict__ ws,
                                      float* __restrict__ out) {
    __shared__ float red[NT];
    float s = 0.0f;
    for (int i = threadIdx.x; i < NB; i += NT) s += ws[i];
    red[threadIdx.x] = s;
    __syncthreads();
    for (int k = NT / 2; k > 0; k >>= 1) {   // fixed tree: deterministic
        if (threadIdx.x < k) red[threadIdx.x] += red[threadIdx.x + k];
        __syncthreads();
    }
    if (threadIdx.x == 0) out[0] = red[0] * (1.0f / (float)NB);
}

extern "C" void kernel_launch(void* const* d_in, const int* in_sizes, int n_in,
                              void* d_out, int out_size, void* d_ws, size_t ws_size,
                              hipStream_t stream) {
    const float* logits  = (const float*)d_in[0];
    const float* targets = (const float*)d_in[1];
    float* ws  = (float*)d_ws;     // 4096 per-row weighted losses
    float* out = (float*)d_out;    // scalar
    SoftMarginRankLoss_row_kernel<<<NB, NT, 0, stream>>>(logits, targets, ws);
    SoftMarginRankLoss_reduce_kernel<<<1, NT, 0, stream>>>(ws, out);
}